// MultAttnDecoderRNN_85624468013568
// MI455X (gfx1250) — compile-verified
//
#include <hip/hip_runtime.h>
#include <math.h>
#include <stdint.h>

typedef __attribute__((ext_vector_type(2))) float v2f;
typedef __attribute__((ext_vector_type(4))) float v4f;
typedef __attribute__((ext_vector_type(8))) float v8f;
typedef __attribute__((ext_vector_type(4))) unsigned int u32x4;
typedef __attribute__((ext_vector_type(8))) unsigned int u32x8;

#define H 1024
#define V 50257
#define S 4096

#define NSPLIT_GI 8
#define NSPLIT_GH 4
#define NSPLIT_SC 4

// ---------------------------------------------------------------------------
// f32 GEMV via V_WMMA_F32_16X16X4_F32 with K-split partials.
// The shared activation vector (reused by all 8 waves every iteration) is
// staged into LDS by the Tensor Data Mover (1x K tile, 2D descriptor,
// groups 2/3 NULL), issued by wave 0 and fenced with s_wait_tensorcnt.
// Weights are streamed global->VGPR directly: in a GEMV each weight is used
// exactly once, so an LDS hop would only add on-chip traffic.
//
// Partial out[blockIdx.y*rows + r] = sum_{k in slice} W[r*K+k]*vec[k]
//                                    (+ bias[r] on slice 0 if bias != null)
// Per outer iteration (8 K-values, 2 WMMAs):
//   lane-half 0 loads float4 W[row][k..k+3]   -> A slots {0,1} of wmma0/wmma1
//   lane-half 1 loads float4 W[row][k+4..k+7] -> A slots {2,3} of wmma0/wmma1
//   B column 0 (lanes with N==0) carries the matching vec values, rest zero.
// D column 0: lane 0 holds M=0..7 in c[0..7]; lane 16 holds M=8..15.
// ---------------------------------------------------------------------------
__global__ __launch_bounds__(256) void gemv_wmma_f32(
    const float* __restrict__ W, const float* __restrict__ vec,
    const float* __restrict__ bias, float* __restrict__ out,
    int rows, int K, int kchunk)
{
    __shared__ float sv[2048];

    if (threadIdx.x < 32) {   // wave 0 issues the TDM DMA (EXEC ignored by TDM)
        unsigned lds_off = (unsigned)(uintptr_t)(&sv[0]);   // addr[31:0] = LDS byte offset
        unsigned long long ga = (unsigned long long)(uintptr_t)vec;
        unsigned uk = (unsigned)K;

        u32x4 g0;
        g0[0] = 1u;                                         // count=1, user descriptor
        g0[1] = lds_off;                                    // lds_addr
        g0[2] = (unsigned)(ga & 0xFFFFFFFFu);               // global_addr[31:0]
        g0[3] = (unsigned)((ga >> 32) & 0x01FFFFFFu)        // global_addr[56:32]
              | (2u << 30);                                 // type = 2 ("image")

        u32x8 g1;
        g1[0] = 0x00020000u;                                // data_size = 2 -> 4 bytes
        g1[1] = (uk & 0xFFFFu) << 16;                       // tensor_dim0[15:0]
        g1[2] = ((uk >> 16) & 0xFFFFu) | (1u << 16);        // tensor_dim0[31:16] | tensor_dim1=1
        g1[3] = (uk & 0xFFFFu) << 16;                       // tile_dim0 = K (<= 2048)
        g1[4] = 1u;                                         // tile_dim1 = 1, tile_dim2 = 0
        g1[5] = uk;                                         // tensor_dim0_stride[31:0] = K
        g1[6] = 0u;
        g1[7] = 0u;

        asm volatile("tensor_load_to_lds %0, %1" :: "s"(g0), "s"(g1) : "memory");
        __builtin_amdgcn_s_wait_tensorcnt(0);
    }
    __syncthreads();

    const int lane = threadIdx.x & 31;
    const int wave = threadIdx.x >> 5;
    const int m    = lane & 15;
    const int half = lane >> 4;                 // 0 -> k..k+3, 1 -> k+4..k+7
    const int rowbase = (blockIdx.x * 8 + wave) * 16;

    const int kbeg = blockIdx.y * kchunk;
    const int kend = kbeg + kchunk;
    float* outp = out + (size_t)blockIdx.y * rows;

    int lrow = rowbase + m;
    if (lrow >= rows) lrow = rows - 1;          // clamp loads; stores guarded below
    const float* wr = W + (size_t)lrow * K + 4 * half;
    const int  ksel = 4 * half;
    const bool lead = (m == 0);                 // lanes holding B column 0

    v8f c = {};
    #pragma unroll 4
    for (int k = kbeg; k < kend; k += 8) {
        if ((k & 31) == 0 && (k + 264) <= kend)
            __builtin_prefetch(wr + k + 256, 0, 0);   // global_prefetch_b8

        v4f a4 = *(const v4f*)(wr + k);               // B128 weight load
        v4f y4 = *(const v4f*)(&sv[k + ksel]);        // B128 LDS read

        v2f a0; a0.x = a4.x; a0.y = a4.y;
        v2f a1; a1.x = a4.z; a1.y = a4.w;
        v2f b0; b0.x = lead ? y4.x : 0.0f; b0.y = lead ? y4.y : 0.0f;
        v2f b1; b1.x = lead ? y4.z : 0.0f; b1.y = lead ? y4.w : 0.0f;

        c = __builtin_amdgcn_wmma_f32_16x16x4_f32(false, a0, false, b0,
                                                  (short)0, c, false, false);
        c = __builtin_amdgcn_wmma_f32_16x16x4_f32(false, a1, false, b1,
                                                  (short)0, c, false, false);
    }

    if (m == 0) {
        #pragma unroll
        for (int v = 0; v < 8; ++v) {
            int r = rowbase + half * 8 + v;
            if (r < rows) {
                float b = (bias && blockIdx.y == 0) ? bias[r] : 0.0f;
                outp[r] = c[v] + b;
            }
        }
    }
}

// x = concat(embedding[word], last_context)
__global__ __launch_bounds__(256) void build_x(
    const int* __restrict__ word, const float* __restrict__ emb,
    const float* __restrict__ last_ctx, float* __restrict__ x)
{
    int i = blockIdx.x * 256 + threadIdx.x;     // grid = 8 blocks -> 2048
    if (i < H)            x[i] = emb[(size_t)word[0] * H + i];
    else if (i < 2 * H)   x[i] = last_ctx[i - H];
}

// GRU gate math over K-split partials; writes h_new into y[0:H] and output.
__global__ __launch_bounds__(256) void gru_gates(
    const float* __restrict__ gi, const float* __restrict__ gh,
    const float* __restrict__ h_in, float* __restrict__ y,
    float* __restrict__ out_hnew)
{
    int j = blockIdx.x * 256 + threadIdx.x;     // grid = 4 blocks -> 1024
    if (j < H) {
        float ir = 0.f, iz = 0.f, in = 0.f;
        #pragma unroll
        for (int s = 0; s < NSPLIT_GI; ++s) {
            const float* g = gi + (size_t)s * 3 * H;
            ir += g[j]; iz += g[H + j]; in += g[2 * H + j];
        }
        float hr = 0.f, hz = 0.f, hn_ = 0.f;
        #pragma unroll
        for (int s = 0; s < NSPLIT_GH; ++s) {
            const float* g = gh + (size_t)s * 3 * H;
            hr += g[j]; hz += g[H + j]; hn_ += g[2 * H + j];
        }
        float r  = 1.0f / (1.0f + expf(-(ir + hr)));
        float z  = 1.0f / (1.0f + expf(-(iz + hz)));
        float n  = tanhf(in + r * hn_);
        float hn = (1.0f - z) * n + z * h_in[j];
        y[j] = hn;
        out_hnew[j] = hn;
    }
}

// Sums NSPLIT_SC score partials into scores_c, then softmax over 4096.
__global__ __launch_bounds__(1024) void softmax4096(
    const float* __restrict__ scores_p, float* __restrict__ scores_c,
    float* __restrict__ attn_ws, float* __restrict__ out_attn)
{
    __shared__ float red[1024];
    int t = threadIdx.x;
    float mx = -1e30f;
    for (int i = t; i < S; i += 1024) {
        float sc = 0.f;
        #pragma unroll
        for (int s = 0; s < NSPLIT_SC; ++s) sc += scores_p[(size_t)s * S + i];
        scores_c[i] = sc;
        mx = fmaxf(mx, sc);
    }
    red[t] = mx; __syncthreads();
    for (int s = 512; s > 0; s >>= 1) { if (t < s) red[t] = fmaxf(red[t], red[t + s]); __syncthreads(); }
    float gmax = red[0]; __syncthreads();
    float sum = 0.0f;
    for (int i = t; i < S; i += 1024) { float e = expf(scores_c[i] - gmax); attn_ws[i] = e; sum += e; }
    red[t] = sum; __syncthreads();
    for (int s = 512; s > 0; s >>= 1) { if (t < s) red[t] += red[t + s]; __syncthreads(); }
    float inv = 1.0f / red[0];
    for (int i = t; i < S; i += 1024) { float a = attn_ws[i] * inv; attn_ws[i] = a; out_attn[i] = a; }
}

// context[j] = sum_i attn[i] * enc[i*H + j]; writes y[H + j] and context output.
__global__ __launch_bounds__(256) void attn_context(
    const float* __restrict__ attn, const float* __restrict__ enc,
    float* __restrict__ y, float* __restrict__ out_ctx)
{
    __shared__ float sa[256];
    int j = blockIdx.x * 256 + threadIdx.x;     // grid = 4 blocks -> 1024 cols
    float acc = 0.0f;
    for (int i0 = 0; i0 < S; i0 += 256) {
        __syncthreads();
        sa[threadIdx.x] = attn[i0 + threadIdx.x];
        __syncthreads();
        #pragma unroll 8
        for (int ii = 0; ii < 256; ++ii)
            acc += sa[ii] * enc[(size_t)(i0 + ii) * H + j];
    }
    y[H + j]   = acc;
    out_ctx[j] = acc;
}

__global__ __launch_bounds__(1024) void logsoftmaxV(
    const float* __restrict__ logits, float* __restrict__ out)
{
    __shared__ float red[1024];
    int t = threadIdx.x;
    float mx = -1e30f;
    for (int i = t; i < V; i += 1024) mx = fmaxf(mx, logits[i]);
    red[t] = mx; __syncthreads();
    for (int s = 512; s > 0; s >>= 1) { if (t < s) red[t] = fmaxf(red[t], red[t + s]); __syncthreads(); }
    float gmax = red[0]; __syncthreads();
    float sum = 0.0f;
    for (int i = t; i < V; i += 1024) sum += expf(logits[i] - gmax);
    red[t] = sum; __syncthreads();
    for (int s = 512; s > 0; s >>= 1) { if (t < s) red[t] += red[t + s]; __syncthreads(); }
    float lse = gmax + logf(red[0]);
    for (int i = t; i < V; i += 1024) out[i] = logits[i] - lse;
}

extern "C" void kernel_launch(void* const* d_in, const int* in_sizes, int n_in,
                              void* d_out, int out_size, void* d_ws, size_t ws_size,
                              hipStream_t stream)
{
    const int*   word         = (const int*)  d_in[0];
    const float* last_context = (const float*)d_in[1];
    const float* last_hidden  = (const float*)d_in[2];
    const float* enc          = (const float*)d_in[3];   // (S,1,H) contiguous
    const float* embedding    = (const float*)d_in[4];
    const float* w_ih         = (const float*)d_in[5];
    const float* w_hh         = (const float*)d_in[6];
    const float* b_ih         = (const float*)d_in[7];
    const float* b_hh         = (const float*)d_in[8];
    const float* out_w        = (const float*)d_in[9];
    const float* out_b        = (const float*)d_in[10];

    float* out        = (float*)d_out;
    float* out_output = out;                 // V
    float* out_ctx    = out + V;             // H
    float* out_hnew   = out + V + H;         // H
    float* out_attn   = out + V + 2 * H;     // S

    float* ws       = (float*)d_ws;
    float* x        = ws;                    // 2048
    float* gi       = ws + 2048;             // 8 * 3072 = 24576
    float* gh       = ws + 26624;            // 4 * 3072 = 12288
    float* scores_p = ws + 38912;            // 4 * 4096 = 16384
    float* scores_c = ws + 55296;            // 4096
    float* attn     = ws + 59392;            // 4096
    float* y        = ws + 63488;            // 2048 = concat(h_new, context)
    float* logits   = ws + 65536;            // 50304

    build_x<<<8, 256, 0, stream>>>(word, embedding, last_context, x);

    // bias applied on slice 0 only (blockIdx.y == 0) inside gemv_wmma_f32
    gemv_wmma_f32<<<dim3(24, NSPLIT_GI), 256, 0, stream>>>(
        w_ih, x, b_ih, gi, 3 * H, 2 * H, (2 * H) / NSPLIT_GI);
    gemv_wmma_f32<<<dim3(24, NSPLIT_GH), 256, 0, stream>>>(
        w_hh, last_hidden, b_hh, gh, 3 * H, H, H / NSPLIT_GH);
    gru_gates<<<4, 256, 0, stream>>>(gi, gh, last_hidden, y, out_hnew);

    gemv_wmma_f32<<<dim3(32, NSPLIT_SC), 256, 0, stream>>>(
        enc, y, nullptr, scores_p, S, H, H / NSPLIT_SC);
    softmax4096<<<1, 1024, 0, stream>>>(scores_p, scores_c, attn, out_attn);
    attn_context<<<4, 256, 0, stream>>>(attn, enc, y, out_ctx);

    gemv_wmma_f32<<<dim3((V + 127) / 128, 1), 256, 0, stream>>>(
        out_w, y, out_b, logits, V, 2 * H, 2 * H);
    logsoftmaxV<<<1, 1024, 0, stream>>>(logits, out_output);
}